// GraphSAGE_5428838662375
// MI455X (gfx1250) — compile-verified
//
#include <hip/hip_runtime.h>
#include <hip/hip_bf16.h>

typedef __attribute__((ext_vector_type(2))) float v2f;
typedef __attribute__((ext_vector_type(8))) float v8f;

#define D_FEAT 128
#define LDS_PAD 4
#define LDS_STRIDE (D_FEAT + LDS_PAD)

// ---------------------------------------------------------------------------
// Zero a float buffer with float4 stores.
__global__ void __launch_bounds__(256)
zero_f32_kernel(float* __restrict__ p, long long n4) {
    long long i = (long long)blockIdx.x * blockDim.x + threadIdx.x;
    if (i < n4) ((float4*)p)[i] = make_float4(0.f, 0.f, 0.f, 0.f);
}

// ---------------------------------------------------------------------------
// Edge-parallel scatter-add: one wave (32 lanes x float4) per edge.
// msg[dst] += feat[src]; optionally deg[dst] += 1.
__global__ void __launch_bounds__(256)
scatter_kernel(const float* __restrict__ feat,
               const int* __restrict__ src,
               const int* __restrict__ dst,
               float* __restrict__ msg,
               float* __restrict__ deg,
               int n_edges, int do_deg)
{
    long long t = (long long)blockIdx.x * blockDim.x + threadIdx.x;
    int e = (int)(t >> 5);
    int c = (int)(t & 31);
    if (e >= n_edges) return;
    int s = src[e];
    int d = dst[e];
    const float4 v = *(const float4*)(feat + (long long)s * D_FEAT + c * 4);
    float* p = msg + (long long)d * D_FEAT + c * 4;
    unsafeAtomicAdd(p + 0, v.x);
    unsafeAtomicAdd(p + 1, v.y);
    unsafeAtomicAdd(p + 2, v.z);
    unsafeAtomicAdd(p + 3, v.w);
    if (do_deg && c == 0) unsafeAtomicAdd(deg + d, 1.0f);
}

// ---------------------------------------------------------------------------
// Fused SAGE layer GEMM:
//   out[i,:] = act( (agg[i,:]/max(deg[i],1)) @ wl^T + bl + xin[i,:] @ wr^T )
// Block: 256 threads = 8 waves; block handles 16 nodes x 128 outputs.
// Wave w computes the 16x16 output block at columns [16w, 16w+16).
// Single fused K-loop with TWO independent WMMA accumulator chains
// (mean@wl^T -> acc0, x@wr^T -> acc1) so the XDL pipe can interleave them.
//
// Fragment layout (ISA 7.12.2):
//   A 16x4 : lane L -> row M=L%16; VGPR0/1 = K {0,1} (lanes 0-15), {2,3} (16-31)
//   B 4x16 : lane L -> col N=L%16; VGPR0/1 = K {0,1} (lanes 0-15), {2,3} (16-31)
//   C/D    : lane L -> col N=L%16; VGPR r = row r (lanes 0-15) / r+8 (16-31)
__global__ void __launch_bounds__(256, 2)
sage_gemm_kernel(const float* __restrict__ agg,
                 const float* __restrict__ deg,
                 const float* __restrict__ xin,
                 const float* __restrict__ wl,
                 const float* __restrict__ bl,
                 const float* __restrict__ wr,
                 float* __restrict__ out,
                 int relu)
{
    __shared__ float lds_mean[16 * LDS_STRIDE];
    __shared__ float lds_x[16 * LDS_STRIDE];

    const int tid = threadIdx.x;
    const long long row0 = (long long)blockIdx.x * 16;   // n_nodes % 16 == 0

    // Cooperative tile staging: each thread handles 8 floats of each matrix.
    {
        int r  = tid >> 4;           // 0..15  node row in tile
        int c0 = (tid & 15) * 8;     // 0..120 column start
        float rdeg = 1.0f / fmaxf(deg[row0 + r], 1.0f);
        const float4* ag = (const float4*)(agg + (row0 + r) * D_FEAT + c0);
        const float4* xg = (const float4*)(xin + (row0 + r) * D_FEAT + c0);
        float4 a0 = ag[0], a1 = ag[1];
        float4 x0 = xg[0], x1 = xg[1];
        float* lm = lds_mean + r * LDS_STRIDE + c0;
        float* lx = lds_x    + r * LDS_STRIDE + c0;
        lm[0] = a0.x * rdeg; lm[1] = a0.y * rdeg;
        lm[2] = a0.z * rdeg; lm[3] = a0.w * rdeg;
        lm[4] = a1.x * rdeg; lm[5] = a1.y * rdeg;
        lm[6] = a1.z * rdeg; lm[7] = a1.w * rdeg;
        *(float4*)(lx)     = x0;
        *(float4*)(lx + 4) = x1;
    }
    __syncthreads();

    const int wave  = tid >> 5;            // 0..7  -> output column block
    const int lane  = tid & 31;
    const int m     = lane & 15;           // A row / B,C,D column within block
    const int khalf = (lane < 16) ? 0 : 2; // K sub-pair for this lane half
    const int n     = wave * 16 + m;       // global output column

    const float bias = bl[n];
    v8f acc0, acc1;
#pragma unroll
    for (int r = 0; r < 8; ++r) { acc0[r] = bias; acc1[r] = 0.0f; }

    const float* am    = lds_mean + m * LDS_STRIDE + khalf;
    const float* ax    = lds_x    + m * LDS_STRIDE + khalf;
    const float* wlrow = wl + (long long)n * D_FEAT + khalf;
    const float* wrrow = wr + (long long)n * D_FEAT + khalf;

    // Fused K loop: two independent accumulation chains.
#pragma unroll 4
    for (int k0 = 0; k0 < D_FEAT; k0 += 4) {
        v2f a0 = *(const v2f*)(am + k0);
        v2f b0 = *(const v2f*)(wlrow + k0);
        v2f a1 = *(const v2f*)(ax + k0);
        v2f b1 = *(const v2f*)(wrrow + k0);
        acc0 = __builtin_amdgcn_wmma_f32_16x16x4_f32(
            false, a0, false, b0, (short)0, acc0, false, false);
        acc1 = __builtin_amdgcn_wmma_f32_16x16x4_f32(
            false, a1, false, b1, (short)0, acc1, false, false);
    }

    v8f res;
#pragma unroll
    for (int r = 0; r < 8; ++r) res[r] = acc0[r] + acc1[r];
    if (relu) {
#pragma unroll
        for (int r = 0; r < 8; ++r) res[r] = fmaxf(res[r], 0.0f);
    }

    const int mbase = (lane < 16) ? 0 : 8;
#pragma unroll
    for (int r = 0; r < 8; ++r) {
        out[(row0 + mbase + r) * D_FEAT + n] = res[r];
    }
}

// ---------------------------------------------------------------------------
extern "C" void kernel_launch(void* const* d_in, const int* in_sizes, int n_in,
                              void* d_out, int out_size, void* d_ws, size_t ws_size,
                              hipStream_t stream) {
    const float* x    = (const float*)d_in[0];
    const float* w1_l = (const float*)d_in[1];
    const float* b1_l = (const float*)d_in[2];
    const float* w1_r = (const float*)d_in[3];
    const float* w2_l = (const float*)d_in[4];
    const float* b2_l = (const float*)d_in[5];
    const float* w2_r = (const float*)d_in[6];
    const int*   edge = (const int*)d_in[7];

    const int n_nodes = in_sizes[0] / D_FEAT;   // 100000 (multiple of 16)
    const int n_edges = in_sizes[7] / 2;        // 1600000
    const int* src = edge;
    const int* dst = edge + n_edges;

    // Workspace layout: msg_sum [N*128] | deg [N] | h [N*128]
    float* msg  = (float*)d_ws;
    float* deg  = msg + (long long)n_nodes * D_FEAT;
    float* h    = deg + n_nodes;
    float* outp = (float*)d_out;

    const long long msg_elems = (long long)n_nodes * D_FEAT;

    // --- Layer 1 ---
    {   // zero msg_sum + deg (contiguous)
        long long n4 = (msg_elems + n_nodes + 3) / 4;
        int grid = (int)((n4 + 255) / 256);
        zero_f32_kernel<<<grid, 256, 0, stream>>>(msg, n4);
    }
    {   // scatter-add features + degrees
        long long threads = (long long)n_edges * 32;
        int grid = (int)((threads + 255) / 256);
        scatter_kernel<<<grid, 256, 0, stream>>>(x, src, dst, msg, deg, n_edges, 1);
    }
    {   // h = relu(mean @ w1_l^T + b1_l + x @ w1_r^T)
        int grid = n_nodes / 16;
        sage_gemm_kernel<<<grid, 256, 0, stream>>>(msg, deg, x, w1_l, b1_l, w1_r, h, 1);
    }

    // --- Layer 2 ---
    {   // zero msg_sum only (deg reused)
        long long n4 = (msg_elems + 3) / 4;
        int grid = (int)((n4 + 255) / 256);
        zero_f32_kernel<<<grid, 256, 0, stream>>>(msg, n4);
    }
    {
        long long threads = (long long)n_edges * 32;
        int grid = (int)((threads + 255) / 256);
        scatter_kernel<<<grid, 256, 0, stream>>>(h, src, dst, msg, deg, n_edges, 0);
    }
    {   // out = mean2 @ w2_l^T + b2_l + h @ w2_r^T
        int grid = n_nodes / 16;
        sage_gemm_kernel<<<grid, 256, 0, stream>>>(msg, deg, h, w2_l, b2_l, w2_r, outp, 0);
    }
}